// BaseMessagePassing_36026185679311
// MI455X (gfx1250) — compile-verified
//
#include <hip/hip_runtime.h>
#include <math.h>

#define HID    256
#define H2     128
#define NB     32000      // B * N_NODES
#define NE     512000
#define NCOMBO 624        // 39 etypes * 4 head * 4 tail
#define MTOT   544000.0f  // NE + NB rows in batchnorm

typedef float v8f __attribute__((ext_vector_type(8)));
typedef float v2f __attribute__((ext_vector_type(2)));

__device__ __forceinline__ float gelu_f(float x) {
  // jax.nn.gelu default (approximate=True, tanh form)
  return 0.5f * x * (1.0f + tanhf(0.7978845608028654f * (x + 0.044715f * x * x * x)));
}
__device__ __forceinline__ float4 f4add(const float4& a, const float4& b) {
  return make_float4(a.x + b.x, a.y + b.y, a.z + b.z, a.w + b.w);
}

// ---------------------------------------------------------------- init
__global__ __launch_bounds__(256) void k_init(unsigned* deg, unsigned* cnt) {
  int i = blockIdx.x * 256 + threadIdx.x;
  if (i < NB) deg[i] = 0u;
  else if (i < NB + NCOMBO) cnt[i - NB] = 0u;
}

// ------------------------------------------- xsum = x + x_extra  [NB,256]
// first 128 ch: gelu(emb_node_type[type]); last 128: gelu(sin(js*score)@W_score+b)
__global__ __launch_bounds__(256) void k_xsum(
    const float* __restrict__ nf, const float* __restrict__ score,
    const int* __restrict__ ntype, const float* __restrict__ emb_nt,
    const float* __restrict__ Wsc, const float* __restrict__ bsc,
    float* __restrict__ xsum) {
  __shared__ float sEnc[2][H2];
  const int t  = threadIdx.x;
  const int ln = t >> 7;          // sub-block node (0/1)
  const int c  = t & 127;
  const float js = __expf((float)c * 0.09531017980432486f); // 1.1^c
  const int base = blockIdx.x * 32;
  for (int step = 0; step < 16; ++step) {
    const int node = base + step * 2 + ln;
    sEnc[ln][c] = __sinf(js * score[node]);
    __syncthreads();
    float acc = bsc[c];
#pragma unroll 4
    for (int k = 0; k < H2; ++k) acc = fmaf(sEnc[ln][k], Wsc[k * H2 + c], acc);
    const float se = gelu_f(acc);
    const float ne = gelu_f(emb_nt[ntype[node] * H2 + c]);
    const float* nfr = nf + (size_t)node * HID;
    float* xr = xsum + (size_t)node * HID;
    xr[c]      = nfr[c] + ne;
    xr[H2 + c] = nfr[H2 + c] + se;
    __syncthreads();
  }
}

// ------------------------------- combo table: h = We1[et]+We1[39+h]+We1[43+t]+b
__global__ __launch_bounds__(256) void k_combo_h(
    const float* __restrict__ We1, const float* __restrict__ be1,
    float* __restrict__ combo_h) {
  const int q = blockIdx.x, c = threadIdx.x;
  const int et = q >> 4, hv = (q >> 2) & 3, tv = q & 3;
  combo_h[q * HID + c] = We1[et * HID + c] + We1[(39 + hv) * HID + c] +
                         We1[(43 + tv) * HID + c] + be1[c];
}

// -------------------- per-edge combo id + dst-degree + combo histogram
__global__ __launch_bounds__(256) void k_edge_pre(
    const int* __restrict__ eidx, const int* __restrict__ etype,
    const int* __restrict__ ntype, int* __restrict__ combo_arr,
    unsigned* deg, unsigned* cnt) {
  const int i = blockIdx.x * 256 + threadIdx.x;
  if (i < NE) {
    const int s = eidx[i], d = eidx[NE + i];
    const int q = etype[i] * 16 + ntype[s] * 4 + ntype[d];
    combo_arr[i] = q;
    atomicAdd(&deg[d], 1u);
    atomicAdd(&cnt[q], 1u);
  } else if (i < NE + NB) {
    const int n = i - NE;                   // self-loop row in BN stats
    atomicAdd(&cnt[608 + 5 * ntype[n]], 1u); // etype=38, head=tail=type
  }
}

// -------------------------------- exclusive scan of deg -> offs, cursor
__global__ __launch_bounds__(256) void k_scan(const unsigned* __restrict__ deg,
                                              unsigned* offs, unsigned* cursor) {
  __shared__ unsigned part[256];
  const int t = threadIdx.x, b0 = t * 125;   // 256*125 = 32000
  unsigned s = 0;
  for (int i = 0; i < 125; ++i) s += deg[b0 + i];
  part[t] = s;
  __syncthreads();
  for (int off = 1; off < 256; off <<= 1) {
    unsigned v = (t >= off) ? part[t - off] : 0u;
    __syncthreads();
    part[t] += v;
    __syncthreads();
  }
  unsigned run = (t == 0) ? 0u : part[t - 1];
  for (int i = 0; i < 125; ++i) {
    const unsigned d = deg[b0 + i];
    offs[b0 + i] = run;
    cursor[b0 + i] = run;
    run += d;
  }
  if (t == 255) offs[NB] = run;
}

// ------------------------------------------- fill CSR: entry = combo<<15 | src
__global__ __launch_bounds__(256) void k_csr(const int* __restrict__ eidx,
                                             const int* __restrict__ combo_arr,
                                             unsigned* cursor, unsigned* csr) {
  const int e = blockIdx.x * 256 + threadIdx.x;
  if (e >= NE) return;
  const unsigned d = (unsigned)eidx[NE + e];
  const unsigned pos = atomicAdd(&cursor[d], 1u);
  csr[pos] = ((unsigned)combo_arr[e] << 15) | (unsigned)eidx[e];
}

// ---------------- BN stats from 624-bin histogram -> fused scale/shift
__global__ __launch_bounds__(256) void k_bnstats(
    const float* __restrict__ combo_h, const unsigned* __restrict__ cnt,
    const float* __restrict__ gam, const float* __restrict__ bet,
    float* scale, float* shift) {
  const int c = threadIdx.x;
  float s1 = 0.f, s2 = 0.f;
  for (int q = 0; q < NCOMBO; ++q) {
    const float h = combo_h[q * HID + c];
    const float w = (float)cnt[q];
    s1 = fmaf(w, h, s1);
    s2 = fmaf(w * h, h, s2);
  }
  const float mu  = s1 / MTOT;
  const float var = s2 / MTOT - mu * mu;
  const float sc  = gam[c] * rsqrtf(var + 1e-5f);
  scale[c] = sc;
  shift[c] = bet[c] - mu * sc;
}

// ------------- combo_emb = relu(bn(h)) @ W_e2 + b_e2   [624,256]
__global__ __launch_bounds__(256) void k_combo_emb(
    const float* __restrict__ combo_h, const float* __restrict__ scale,
    const float* __restrict__ shift, const float* __restrict__ We2,
    const float* __restrict__ be2, float* __restrict__ combo_emb) {
  __shared__ float r[HID];
  const int q = blockIdx.x, t = threadIdx.x;
  r[t] = fmaxf(fmaf(combo_h[q * HID + t], scale[t], shift[t]), 0.f);
  __syncthreads();
  float acc = be2[t];
#pragma unroll 4
  for (int k = 0; k < HID; ++k) acc = fmaf(r[k], We2[k * HID + t], acc);
  combo_emb[q * HID + t] = acc;
}

// ----- aggregation: one wave per dst node, register accumulators, no atomics
__global__ __launch_bounds__(256) void k_agg(
    const float* __restrict__ xsum, const float* __restrict__ combo_emb,
    const unsigned* __restrict__ offs, const unsigned* __restrict__ csr,
    const int* __restrict__ ntype, float* __restrict__ agg) {
  const int wave = threadIdx.x >> 5, lane = threadIdx.x & 31;
  const int d = blockIdx.x * 8 + wave;
  const float4* xs = (const float4*)xsum;
  const float4* ce = (const float4*)combo_emb;
  // self-loop contribution: xsum[d] + combo_emb[self]
  const int sq = 608 + 5 * ntype[d];
  float4 a0 = f4add(xs[(size_t)d * 64 + lane],      ce[(size_t)sq * 64 + lane]);
  float4 a1 = f4add(xs[(size_t)d * 64 + 32 + lane], ce[(size_t)sq * 64 + 32 + lane]);
  const unsigned beg = offs[d], end = offs[d + 1];
  for (unsigned p = beg; p != end; ++p) {
    const unsigned ent = csr[p];
    const size_t s = (size_t)(ent & 0x7FFFu);
    const size_t q = (size_t)(ent >> 15);
    a0 = f4add(a0, f4add(xs[s * 64 + lane],      ce[q * 64 + lane]));
    a1 = f4add(a1, f4add(xs[s * 64 + 32 + lane], ce[q * 64 + 32 + lane]));
  }
  ((float4*)agg)[(size_t)d * 64 + lane]      = a0;
  ((float4*)agg)[(size_t)d * 64 + 32 + lane] = a1;
}

// ---------------- WT[n][k] = concat(W_vh, W_vx) column-major  [256][512]
__global__ __launch_bounds__(256) void k_wt(const float* __restrict__ Wvh,
                                            const float* __restrict__ Wvx,
                                            float* __restrict__ WT) {
  const int i = blockIdx.x * 256 + threadIdx.x; // 0..131071
  const int n = i >> 9, k = i & 511;
  WT[i] = (k < HID) ? Wvh[k * HID + n] : Wvx[(k - HID) * HID + n];
}

// ---- out = gelu([NF | AGG] @ WT^T + bvh + bvx) via V_WMMA_F32_16X16X4_F32
__global__ __launch_bounds__(256) void k_out(
    const float* __restrict__ nf, const float* __restrict__ agg,
    const float* __restrict__ WT, const float* __restrict__ bvh,
    const float* __restrict__ bvx, float* __restrict__ out) {
  __shared__ float sA[16 * 516];               // 16 rows x 512 K, pad 4 (33 KB)
  const int t = threadIdx.x;
  const int row0 = blockIdx.x * 16;
  // stage A strip (coalesced float4; 516*4 bytes keeps rows 16B-aligned)
#pragma unroll
  for (int i = 0; i < 8; ++i) {
    const int flat = t + 256 * i;              // float4 slot 0..2047
    const int r = flat >> 7;
    const int c = (flat & 127) << 2;
    const float* src = (c < HID) ? (nf + (size_t)(row0 + r) * HID + c)
                                 : (agg + (size_t)(row0 + r) * HID + (c - HID));
    *(float4*)&sA[r * 516 + c] = *(const float4*)src;
  }
  __syncthreads();
  const int wave = t >> 5, lane = t & 31;
  const int hi = lane >> 4, l15 = lane & 15;   // A: lane=M+16*(K/2); B: lane=N+16*(K/2)
  v8f acc0 = {}, acc1 = {};
  const float* arow  = &sA[l15 * 516 + 2 * hi];
  const int n0 = wave * 32;
  const float* brow0 = WT + (size_t)(n0 + l15) * 512 + 2 * hi;
  const float* brow1 = WT + (size_t)(n0 + 16 + l15) * 512 + 2 * hi;
#pragma unroll 4
  for (int k0 = 0; k0 < 512; k0 += 4) {
    const v2f a  = *(const v2f*)(arow + k0);
    const v2f b0 = *(const v2f*)(brow0 + k0);
    const v2f b1 = *(const v2f*)(brow1 + k0);
    acc0 = __builtin_amdgcn_wmma_f32_16x16x4_f32(false, a, false, b0, (short)0,
                                                 acc0, false, false);
    acc1 = __builtin_amdgcn_wmma_f32_16x16x4_f32(false, a, false, b1, (short)0,
                                                 acc1, false, false);
  }
  // D layout: vgpr r -> M=r (lanes 0-15) / M=r+8 (lanes 16-31); lane&15 = N
#pragma unroll
  for (int tt = 0; tt < 2; ++tt) {
    const int n = n0 + tt * 16 + l15;
    const float bias = bvh[n] + bvx[n];
    const v8f acc = tt ? acc1 : acc0;
#pragma unroll
    for (int r = 0; r < 8; ++r) {
      const int m = r + 8 * hi;
      out[(size_t)(row0 + m) * HID + n] = gelu_f(acc[r] + bias);
    }
  }
}

extern "C" void kernel_launch(void* const* d_in, const int* in_sizes, int n_in,
                              void* d_out, int out_size, void* d_ws,
                              size_t ws_size, hipStream_t stream) {
  const float* nf     = (const float*)d_in[0];
  const int*   eidx   = (const int*)d_in[1];
  const int*   etype  = (const int*)d_in[2];
  const int*   ntype  = (const int*)d_in[3];
  const float* nscore = (const float*)d_in[4];
  const float* emb_nt = (const float*)d_in[5];
  const float* Wsc    = (const float*)d_in[6];
  const float* bsc    = (const float*)d_in[7];
  const float* We1    = (const float*)d_in[8];
  const float* be1    = (const float*)d_in[9];
  const float* gam    = (const float*)d_in[10];
  const float* bet    = (const float*)d_in[11];
  const float* We2    = (const float*)d_in[12];
  const float* be2    = (const float*)d_in[13];
  const float* Wvh    = (const float*)d_in[14];
  const float* bvh    = (const float*)d_in[15];
  const float* Wvx    = (const float*)d_in[16];
  const float* bvx    = (const float*)d_in[17];
  float* out = (float*)d_out;

  char* w = (char*)d_ws;
  size_t off = 0;
  auto take = [&](size_t bytes) -> void* {
    void* p = (void*)(w + off);
    off += (bytes + 255) & ~(size_t)255;
    return p;
  };
  float*    xsum      = (float*)take((size_t)NB * HID * 4);
  float*    agg       = (float*)take((size_t)NB * HID * 4);
  float*    combo_h   = (float*)take((size_t)NCOMBO * HID * 4);
  float*    combo_emb = (float*)take((size_t)NCOMBO * HID * 4);
  float*    scale     = (float*)take(HID * 4);
  float*    shift     = (float*)take(HID * 4);
  unsigned* cnt       = (unsigned*)take(NCOMBO * 4);
  unsigned* deg       = (unsigned*)take((size_t)NB * 4);
  unsigned* offs      = (unsigned*)take((size_t)(NB + 1) * 4);
  unsigned* cursor    = (unsigned*)take((size_t)NB * 4);
  int*      combo_arr = (int*)take((size_t)NE * 4);
  unsigned* csr       = (unsigned*)take((size_t)NE * 4);
  float*    WT        = (float*)take((size_t)512 * HID * 4);

  k_init<<<(NB + NCOMBO + 255) / 256, 256, 0, stream>>>(deg, cnt);
  k_xsum<<<NB / 32, 256, 0, stream>>>(nf, nscore, ntype, emb_nt, Wsc, bsc, xsum);
  k_combo_h<<<NCOMBO, 256, 0, stream>>>(We1, be1, combo_h);
  k_edge_pre<<<(NE + NB + 255) / 256, 256, 0, stream>>>(eidx, etype, ntype,
                                                        combo_arr, deg, cnt);
  k_scan<<<1, 256, 0, stream>>>(deg, offs, cursor);
  k_csr<<<(NE + 255) / 256, 256, 0, stream>>>(eidx, combo_arr, cursor, csr);
  k_bnstats<<<1, 256, 0, stream>>>(combo_h, cnt, gam, bet, scale, shift);
  k_combo_emb<<<NCOMBO, 256, 0, stream>>>(combo_h, scale, shift, We2, be2,
                                          combo_emb);
  k_agg<<<NB / 8, 256, 0, stream>>>(xsum, combo_emb, offs, csr, ntype, agg);
  k_wt<<<(512 * HID) / 256, 256, 0, stream>>>(Wvh, Wvx, WT);
  k_out<<<NB / 16, 256, 0, stream>>>(nf, agg, WT, bvh, bvx, out);
}